// RouteGNN_17257178595365
// MI455X (gfx1250) — compile-verified
//
#include <hip/hip_runtime.h>
#include <hip/hip_bf16.h>
#include <math.h>

typedef __attribute__((ext_vector_type(2))) float v2f;
typedef __attribute__((ext_vector_type(8))) float v8f;

#define TPB 256

// ---------- monotone float<->uint encoding for atomic max ----------
__device__ __forceinline__ unsigned encF(float f) {
    unsigned u = __float_as_uint(f);
    return (u & 0x80000000u) ? ~u : (u | 0x80000000u);
}
__device__ __forceinline__ float decF(unsigned u) {
    return (u & 0x80000000u) ? __uint_as_float(u ^ 0x80000000u) : __uint_as_float(~u);
}

// ---------- utility ----------
__global__ void fill_f32(float* p, float v, int n) {
    int i = blockIdx.x * blockDim.x + threadIdx.x;
    if (i < n) p[i] = v;
}

// zero-pad x [N,5] -> xp [N,8]
__global__ void pad_x_kernel(const float* __restrict__ x, float* __restrict__ xp, int N) {
    int t = blockIdx.x * blockDim.x + threadIdx.x;
    if (t >= N * 8) return;
    int n = t >> 3, k = t & 7;
    xp[t] = (k < 5) ? x[n * 5 + k] : 0.f;
}
// zero-pad W1 [5,128] -> Wp [8,128]
__global__ void pad_w_kernel(const float* __restrict__ W, float* __restrict__ Wp) {
    int t = blockIdx.x * blockDim.x + threadIdx.x;
    if (t >= 8 * 128) return;
    int k = t >> 7;
    Wp[t] = (k < 5) ? W[t] : 0.f;
}

// per-dst edge count and edge_attr sum (for self-loop fill_value='mean')
__global__ void edge_count_kernel(const int* ei, const float* eattr,
                                  float* cnt, float* asum, int E) {
    int e = blockIdx.x * blockDim.x + threadIdx.x;
    if (e >= E) return;
    int dst = ei[E + e];
    atomicAdd(&cnt[dst], 1.0f);
    atomicAdd(&asum[dst], eattr[e]);
}

__global__ void loop_attr_div(float* asum, const float* cnt, int N) {
    int n = blockIdx.x * blockDim.x + threadIdx.x;
    if (n >= N) return;
    asum[n] = asum[n] / fmaxf(cnt[n], 1.0f);
}

// se[h] = dot(We[0, h*64 : h*64+64], ae[h,:])   (edge attention scalar per head)
__global__ void se_kernel(const float* We, const float* ae, float* se) {
    int h = threadIdx.x;
    if (h >= 2) return;
    float s = 0.f;
    for (int c = 0; c < 64; ++c) s += We[h * 64 + c] * ae[h * 64 + c];
    se[h] = s;
}

// ---------- fp32 WMMA GEMM:  C[16*T x 128] = A[.. x K] * B[K x 128] ----------
// Full tiles only — no guards anywhere. Compile-time K (8 or 64).
// One wave per 16x16 tile; 8 waves/block cover the 128-wide output row.
// A lane layout (16x4 f32): lane l -> M=l%16, K=(l/16)*2 + j (j = VGPR 0/1)
// B lane layout (4x16 f32): lane l -> K=(l/16)*2 + j, N=l%16
// D layout: VGPR r, lane l -> row r + (l/16)*8, col l%16
template <int K>
__global__ void wmma_gemm_kernel(const float* __restrict__ A,
                                 const float* __restrict__ B,
                                 float* __restrict__ C) {
    int wave = threadIdx.x >> 5;
    int lane = threadIdx.x & 31;
    int m0 = blockIdx.x * 16;
    int n0 = wave * 16;
    int row = lane & 15;
    int khi = lane >> 4;

    const float* Arow = A + (size_t)(m0 + row) * K + khi * 2;  // 8B aligned
    const float* Bcol = B + (size_t)(khi * 2) * 128 + n0 + row;

    v8f acc = {};
#pragma unroll
    for (int k = 0; k < K; k += 4) {
        v2f a = *(const v2f*)(Arow + k);   // contiguous K pair -> global_load_b64
        v2f b;
        b.x = Bcol[(size_t)k * 128];
        b.y = Bcol[(size_t)k * 128 + 128];
        // 8 args: (neg_a, A, neg_b, B, c_mod, C, reuse_a, reuse_b)
        acc = __builtin_amdgcn_wmma_f32_16x16x4_f32(
            false, a, false, b, (short)0, acc, false, false);
    }
    float* Cp = C + (size_t)(m0 + khi * 8) * 128 + n0 + row;
#pragma unroll
    for (int r = 0; r < 8; ++r)
        Cp[(size_t)r * 128] = acc[r];      // clean clause of 8 b32 stores
}

// scalar fallback for M % 16 remainder rows (not launched when M % 16 == 0)
template <int K>
__global__ void gemm_tail_kernel(const float* __restrict__ A,
                                 const float* __restrict__ B,
                                 float* __restrict__ C, int mStart, int M) {
    int t = blockIdx.x * blockDim.x + threadIdx.x;
    int m = mStart + (t >> 7);
    int c = t & 127;
    if (m >= M) return;
    float s = 0.f;
    for (int k = 0; k < K; ++k) s += A[(size_t)m * K + k] * B[(size_t)k * 128 + c];
    C[(size_t)m * 128 + c] = s;
}

// al_s[n,h], al_d[n,h] = <xh[n,h,:], a_s[h,:]> , <xh[n,h,:], a_d[h,:]>
__global__ void node_attn_kernel(const float* __restrict__ xh,
                                 const float* __restrict__ as_,
                                 const float* __restrict__ ad_,
                                 float* als, float* ald, int N) {
    int n = blockIdx.x * blockDim.x + threadIdx.x;
    if (n >= N) return;
    for (int h = 0; h < 2; ++h) {
        float s = 0.f, d = 0.f;
        const float* xr = xh + (size_t)n * 128 + h * 64;
        for (int c = 0; c < 64; ++c) {
            float v = xr[c];
            s += v * as_[h * 64 + c];
            d += v * ad_[h * 64 + c];
        }
        als[n * 2 + h] = s;
        ald[n * 2 + h] = d;
    }
}

// pass 1: alpha = leaky_relu(al_s[src]+al_d[dst]+eattr*se, 0.2); segment max over dst
__global__ void edge_alpha_kernel(const int* __restrict__ ei,
                                  const float* __restrict__ eattr,
                                  const float* __restrict__ loopattr,
                                  const float* __restrict__ als,
                                  const float* __restrict__ ald,
                                  const float* __restrict__ se,
                                  float* alpha, unsigned* amax, int E, int N) {
    int e = blockIdx.x * blockDim.x + threadIdx.x;
    int Etot = E + N;
    if (e >= Etot) return;
    int src, dst; float ea;
    if (e < E) { src = ei[e]; dst = ei[E + e]; ea = eattr[e]; }
    else       { src = dst = e - E;            ea = loopattr[e - E]; }
    for (int h = 0; h < 2; ++h) {
        float a = als[src * 2 + h] + ald[dst * 2 + h] + ea * se[h];
        a = (a > 0.f) ? a : 0.2f * a;
        alpha[(size_t)e * 2 + h] = a;
        atomicMax(&amax[dst * 2 + h], encF(a));
    }
}

// pass 2: ex = exp(alpha - amax[dst]); den[dst] += ex  (ex overwrites alpha)
__global__ void edge_exp_kernel(const int* __restrict__ ei,
                                const unsigned* __restrict__ amax,
                                float* alpha, float* den, int E, int N) {
    int e = blockIdx.x * blockDim.x + threadIdx.x;
    int Etot = E + N;
    if (e >= Etot) return;
    int dst = (e < E) ? ei[E + e] : (e - E);
    for (int h = 0; h < 2; ++h) {
        float ex = expf(alpha[(size_t)e * 2 + h] - decF(amax[dst * 2 + h]));
        alpha[(size_t)e * 2 + h] = ex;
        atomicAdd(&den[dst * 2 + h], ex);
    }
}

// pass 3: accum[dst,h,c] += xh[src,h,c] * ex/(den[dst,h]+1e-16)
__global__ void edge_scatter_kernel(const int* __restrict__ ei,
                                    const float* __restrict__ xh,
                                    const float* __restrict__ alpha,
                                    const float* __restrict__ den,
                                    float* accum, int E, int N) {
    long long t = (long long)blockIdx.x * blockDim.x + threadIdx.x;
    int Etot = E + N;
    if (t >= (long long)Etot * 128) return;
    int e = (int)(t >> 7);
    int i = (int)(t & 127);
    int src, dst;
    if (e < E) { src = ei[e]; dst = ei[E + e]; }
    else       { src = dst = e - E; }
    int h = i >> 6;
    float att = alpha[(size_t)e * 2 + h] / (den[dst * 2 + h] + 1e-16f);
    atomicAdd(&accum[(size_t)dst * 128 + i], xh[(size_t)src * 128 + i] * att);
}

// h[n,c] = relu( mean over heads + bias )
__global__ void node_out_kernel(const float* __restrict__ accum,
                                const float* __restrict__ b,
                                float* out, int N) {
    int t = blockIdx.x * blockDim.x + threadIdx.x;
    if (t >= N * 64) return;
    int n = t >> 6, c = t & 63;
    float v = 0.5f * (accum[(size_t)n * 128 + c] + accum[(size_t)n * 128 + 64 + c]) + b[c];
    out[(size_t)n * 64 + c] = fmaxf(v, 0.f);
}

// ---------- pooling + final linear ----------
__global__ void pool_count_kernel(const int* batch, float* cntb, int N) {
    int n = blockIdx.x * blockDim.x + threadIdx.x;
    if (n >= N) return;
    atomicAdd(&cntb[batch[n]], 1.0f);
}
__global__ void pool_sum_kernel(const int* batch, const float* h, float* pooled, int N) {
    int t = blockIdx.x * blockDim.x + threadIdx.x;
    if (t >= N * 64) return;
    int n = t >> 6, c = t & 63;
    atomicAdd(&pooled[batch[n] * 64 + c], h[(size_t)n * 64 + c]);
}
__global__ void final_kernel(const float* pooled, const float* cntb,
                             const float* Wlin, const float* blin, float* out) {
    int g = threadIdx.x;
    if (g >= 64) return;
    float inv = 1.0f / fmaxf(cntb[g], 1.0f);
    float s = 0.f;
    for (int c = 0; c < 64; ++c) s += pooled[g * 64 + c] * inv * Wlin[c];
    out[g] = s + blin[0];
}

extern "C" void kernel_launch(void* const* d_in, const int* in_sizes, int n_in,
                              void* d_out, int out_size, void* d_ws, size_t ws_size,
                              hipStream_t stream) {
    const float* x     = (const float*)d_in[0];
    const int*   ei    = (const int*)  d_in[1];
    const float* eattr = (const float*)d_in[2];
    const int*   batch = (const int*)  d_in[3];
    const float* W1  = (const float*)d_in[4];
    const float* as1 = (const float*)d_in[5];
    const float* ad1 = (const float*)d_in[6];
    const float* We1 = (const float*)d_in[7];
    const float* ae1 = (const float*)d_in[8];
    const float* b1  = (const float*)d_in[9];
    const float* W2  = (const float*)d_in[10];
    const float* as2 = (const float*)d_in[11];
    const float* ad2 = (const float*)d_in[12];
    const float* We2 = (const float*)d_in[13];
    const float* ae2 = (const float*)d_in[14];
    const float* b2  = (const float*)d_in[15];
    const float* Wlin = (const float*)d_in[16];
    const float* blin = (const float*)d_in[17];

    const int N = in_sizes[0] / 5;   // 50000
    const int E = in_sizes[2];       // 800000 (edge_attr is [E,1])
    const int Etot = E + N;

    // workspace layout (floats)
    float* ws = (float*)d_ws;
    size_t o = 0;
    float*    cnt      = ws + o; o += (size_t)N;
    float*    loopattr = ws + o; o += (size_t)N;
    float*    xh       = ws + o; o += (size_t)N * 128;
    float*    als      = ws + o; o += (size_t)N * 2;
    float*    ald      = ws + o; o += (size_t)N * 2;
    unsigned* amax     = (unsigned*)(ws + o); o += (size_t)N * 2;
    float*    den      = ws + o; o += (size_t)N * 2;
    float*    alpha    = ws + o; o += (size_t)Etot * 2;
    float*    accum    = ws + o; o += (size_t)N * 128;
    float*    hbuf     = ws + o; o += (size_t)N * 64;
    float*    h2buf    = ws + o; o += (size_t)N * 64;
    float*    xpad     = ws + o; o += (size_t)N * 8;
    float*    w1pad    = ws + o; o += 8 * 128;
    float*    pooled   = ws + o; o += 64 * 64;
    float*    cntb     = ws + o; o += 64;
    float*    se       = ws + o; o += 2;
    (void)ws_size; (void)n_in; (void)out_size;

    auto NB = [](long long n) { return (unsigned)((n + TPB - 1) / TPB); };

    // ---- self-loop edge_attr (fill_value='mean') ----
    fill_f32<<<NB(N), TPB, 0, stream>>>(cnt, 0.f, N);
    fill_f32<<<NB(N), TPB, 0, stream>>>(loopattr, 0.f, N);
    edge_count_kernel<<<NB(E), TPB, 0, stream>>>(ei, eattr, cnt, loopattr, E);
    loop_attr_div<<<NB(N), TPB, 0, stream>>>(loopattr, cnt, N);

    // ---- zero-pad layer-1 operands to K=8 ----
    pad_x_kernel<<<NB((long long)N * 8), TPB, 0, stream>>>(x, xpad, N);
    pad_w_kernel<<<NB(8 * 128), TPB, 0, stream>>>(W1, w1pad);

    const int MtilesFull = N / 16;       // full 16-row tiles (3125 for N=50000)
    const int mRem = N - MtilesFull * 16;

    auto run_gemm = [&](auto kTag, const float* A, const float* B) {
        constexpr int K = decltype(kTag)::value;
        if (MtilesFull > 0)
            wmma_gemm_kernel<K><<<MtilesFull, 256, 0, stream>>>(A, B, xh);
        if (mRem > 0)
            gemm_tail_kernel<K><<<NB((long long)mRem * 128), TPB, 0, stream>>>(
                A, B, xh, MtilesFull * 16, N);
    };

    auto run_edges = [&](const float* as_, const float* ad_,
                         const float* We, const float* ae_, const float* bb,
                         float* hout) {
        se_kernel<<<1, 32, 0, stream>>>(We, ae_, se);
        node_attn_kernel<<<NB(N), TPB, 0, stream>>>(xh, as_, ad_, als, ald, N);
        fill_f32<<<NB((long long)N * 2), TPB, 0, stream>>>((float*)amax, 0.f, N * 2); // enc-min
        fill_f32<<<NB((long long)N * 2), TPB, 0, stream>>>(den, 0.f, N * 2);
        fill_f32<<<NB((long long)N * 128), TPB, 0, stream>>>(accum, 0.f, N * 128);
        edge_alpha_kernel<<<NB(Etot), TPB, 0, stream>>>(ei, eattr, loopattr, als, ald, se,
                                                        alpha, amax, E, N);
        edge_exp_kernel<<<NB(Etot), TPB, 0, stream>>>(ei, amax, alpha, den, E, N);
        edge_scatter_kernel<<<NB((long long)Etot * 128), TPB, 0, stream>>>(
            ei, xh, alpha, den, accum, E, N);
        node_out_kernel<<<NB((long long)N * 64), TPB, 0, stream>>>(accum, bb, hout, N);
    };

    // layer 1: K=8 (padded), layer 2: K=64 — both via v_wmma_f32_16x16x4_f32
    run_gemm(std::integral_constant<int, 8>{}, xpad, w1pad);
    run_edges(as1, ad1, We1, ae1, b1, hbuf);
    run_gemm(std::integral_constant<int, 64>{}, hbuf, W2);
    run_edges(as2, ad2, We2, ae2, b2, h2buf);

    // ---- global mean pool + linear ----
    fill_f32<<<1, 64, 0, stream>>>(cntb, 0.f, 64);
    fill_f32<<<NB(64 * 64), TPB, 0, stream>>>(pooled, 0.f, 64 * 64);
    pool_count_kernel<<<NB(N), TPB, 0, stream>>>(batch, cntb, N);
    pool_sum_kernel<<<NB((long long)N * 64), TPB, 0, stream>>>(batch, h2buf, pooled, N);
    final_kernel<<<1, 64, 0, stream>>>(pooled, cntb, Wlin, blin, (float*)d_out);
}